// TriplaneRender_65300682768546
// MI455X (gfx1250) — compile-verified
//
#include <hip/hip_runtime.h>
#include <hip/hip_bf16.h>

typedef __attribute__((ext_vector_type(2)))  float v2f;
typedef __attribute__((ext_vector_type(8)))  float v8f;
typedef __attribute__((ext_vector_type(4)))  unsigned int v4u;
typedef __attribute__((ext_vector_type(8)))  int v8i;
typedef __attribute__((ext_vector_type(4)))  int v4i;

#define NPTS   (2u * 128u * 128u * 64u)   // 2,097,152 points
#define PLANEF (65536u * 16u)             // floats per (plane,batch) feature plane

// ---------------------------------------------------------------------------
// Kernel A: channel-major (C,HF,HF) -> channel-last (HF*HF, C) transpose.
// One TDM tensor_load_to_lds per block pulls a 16x256 f32 tile into LDS
// (16 channel-rows of 256 pixels, global stride 65536 elems), then the block
// writes it back out channel-last with fully coalesced stores.
// ---------------------------------------------------------------------------
__global__ __launch_bounds__(256)
void tp_transpose_tdm(const float* __restrict__ fxy,
                      const float* __restrict__ fyz,
                      const float* __restrict__ fxz,
                      float* __restrict__ wsT)
{
    __shared__ float lds[16 * 256];          // 16 KB tile: [channel][pix]

    const unsigned int pb    = blockIdx.x >> 8;       // 0..5  = plane*2 + b
    const unsigned int tile  = blockIdx.x & 255u;     // 0..255
    const unsigned int pix0  = tile << 8;             // 256 pixels per tile
    const unsigned int plane = pb >> 1;
    const unsigned int b     = pb & 1u;

    const float* src = (plane == 0) ? fxy : (plane == 1) ? fyz : fxz;
    src += b * (16u * 65536u) + pix0;                 // element (c=0, pix0)

    if (threadIdx.x < 32) {   // one wave issues the DMA (TDM ignores EXEC)
        unsigned long long ga = (unsigned long long)(const void*)src;
        unsigned int ldsOff   = (unsigned int)(unsigned long long)(void*)&lds[0];

        // D# group 0: count=1 | lds_addr | global_addr[56:0] | type=2
        v4u g0;
        g0[0] = 1u;
        g0[1] = ldsOff;
        g0[2] = (unsigned int)ga;
        g0[3] = (unsigned int)((ga >> 32) & 0x01FFFFFFu) | 0x80000000u;

        // D# group 1: data_size=2(4B), tensor_dim0=65536, tensor_dim1=16,
        // tile_dim0=256, tile_dim1=16, tensor_dim0_stride=65536
        v8i g1;
        g1[0] = 0x00020000;              // data_size=4B, mask=0, no pad
        g1[1] = 0;                       // tensor_dim0[15:0]=0 (65536 low)
        g1[2] = 0x00100001;              // tensor_dim0[31:16]=1 | tensor_dim1=16
        g1[3] = 0x01000000;              // tile_dim0 = 256
        g1[4] = 16;                      // tile_dim1 = 16, tile_dim2 = 0
        g1[5] = 0x00010000;              // tensor_dim0_stride = 65536
        g1[6] = 0;
        g1[7] = 0;

        v4i g2 = {0, 0, 0, 0};
        v4i g3 = {0, 0, 0, 0};
        v8i g4 = {0, 0, 0, 0, 0, 0, 0, 0};   // unused dim-3/4 / gather state (2D tile)
        __builtin_amdgcn_tensor_load_to_lds(g0, g1, g2, g3, g4, 0);
        __builtin_amdgcn_s_wait_tensorcnt(0);
    }
    __syncthreads();

    // Write out channel-last: uniform base + 32-bit linear offset (coalesced).
    float* dst = wsT + pb * PLANEF + pix0 * 16u;
#pragma unroll
    for (unsigned int k = 0; k < 16; ++k) {
        unsigned int lin = (k << 8) + threadIdx.x;    // 0..4095
        unsigned int c   = lin & 15u;
        unsigned int pix = lin >> 4;
        dst[lin] = lds[c * 256u + pix];
    }
}

// ---------------------------------------------------------------------------
// Kernel B: one wave per point. For each of the 3 planes the 4-corner x 16-
// channel bilinear blend is computed as a 16x4 * 4x16 f32 WMMA:
//   A row0 = in-bounds-masked corner weights (lane0: K0,K1; lane16: K2,K3)
//   B      = corner features (lanes0-15: corners 0/1; lanes16-31: corners 2/3;
//            channel = lane&15)  -> coalesced 64B loads in the channel-last
//            layout.
//   D row0 = 16 interpolated channels, channel c in lane c.
// Point id is scalarized (readfirstlane) so point/depth reads and plane bases
// are SGPR-uniform; all gather offsets are 32-bit VGPRs -> GVS addressing.
// ---------------------------------------------------------------------------
template <bool TP>
__global__ __launch_bounds__(256)
void tp_main(const float* __restrict__ points,
             const float* __restrict__ depth,
             const float* __restrict__ fxy,
             const float* __restrict__ fyz,
             const float* __restrict__ fxz,
             const float* __restrict__ wsT,
             float* __restrict__ outDens,
             float* __restrict__ outCol)
{
    // wave-uniform point id, provably uniform to the compiler
    const unsigned int wv = (unsigned int)
        __builtin_amdgcn_readfirstlane((int)(threadIdx.x >> 5));
    const unsigned int p  = blockIdx.x * 8u + wv;

    const int  lane = (int)(threadIdx.x & 31u);
    const int  ch   = lane & 15;
    const bool hiK  = lane >= 16;         // upper half-wave holds K=2,3 / corners 2,3

    const unsigned int n = p & 63u;
    const unsigned int b = p >> 20;       // / (64*128*128)

    const float x = points[p * 3u + 0u];
    const float y = points[p * 3u + 1u];
    const float z = points[p * 3u + 2u];

    const float d0v   = depth[p];
    const float delta = (n < 63u) ? (depth[p + 1u] - d0v) : 1e10f;

    float densAcc = 0.0f;

#pragma unroll
    for (int plane = 0; plane < 3; ++plane) {
        const float u = (plane == 2) ? y : x;
        const float v = (plane == 0) ? y : z;

        const float* base;
        if (TP) {
            base = wsT + ((unsigned int)plane * 2u + b) * PLANEF;    // (pix, C)
        } else {
            const float* f = (plane == 0) ? fxy : (plane == 1) ? fyz : fxz;
            base = f + b * (16u * 65536u);                           // (C, pix)
        }

        const float ixf = ((u + 1.0f) * 256.0f - 1.0f) * 0.5f;
        const float iyf = ((v + 1.0f) * 256.0f - 1.0f) * 0.5f;
        const float x0 = floorf(ixf), y0 = floorf(iyf);
        const float wx1 = ixf - x0,  wy1 = iyf - y0;
        const float wx0 = 1.0f - wx1, wy0 = 1.0f - wy1;
        const float x1 = x0 + 1.0f,  y1 = y0 + 1.0f;

        // This lane's two corners: (x0, ysel) and (x1, ysel)
        const float ysel = hiK ? y1 : y0;
        const float wys  = hiK ? wy1 : wy0;
        const bool oky  = (ysel >= 0.0f) && (ysel <= 255.0f);
        const bool okx0 = (x0 >= 0.0f) && (x0 <= 255.0f);
        const bool okx1 = (x1 >= 0.0f) && (x1 <= 255.0f);
        const unsigned int xc0 = (unsigned int)(int)fminf(fmaxf(x0, 0.0f), 255.0f);
        const unsigned int xc1 = (unsigned int)(int)fminf(fmaxf(x1, 0.0f), 255.0f);
        const unsigned int yc  = (unsigned int)(int)fminf(fmaxf(ysel, 0.0f), 255.0f);
        const unsigned int pixA = yc * 256u + xc0;
        const unsigned int pixB = yc * 256u + xc1;

        v2f bm;
        if (TP) {   // channel-last: 16 lanes hit one 64B line per corner
            bm.x = base[pixA * 16u + (unsigned int)ch];
            bm.y = base[pixB * 16u + (unsigned int)ch];
        } else {    // fallback: channel-major strided gather
            bm.x = base[(unsigned int)ch * 65536u + pixA];
            bm.y = base[(unsigned int)ch * 65536u + pixB];
        }

        const float wa0 = (okx0 && oky) ? (wx0 * wys) : 0.0f;
        const float wa1 = (okx1 && oky) ? (wx1 * wys) : 0.0f;
        v2f am;
        am.x = (ch == 0) ? wa0 : 0.0f;    // A[0,K0] in lane0, A[0,K2] in lane16
        am.y = (ch == 0) ? wa1 : 0.0f;    // A[0,K1] in lane0, A[0,K3] in lane16

        v8f cm = {};
        v8f dm = __builtin_amdgcn_wmma_f32_16x16x4_f32(
            false, am, false, bm, (short)0, cm, false, false);

        const float val = dm[0];          // row M=0: lanes 0-15 hold channel=lane
        if (!hiK)
            outCol[p * 48u + (unsigned int)plane * 16u + (unsigned int)ch] = val;

        // channel 0 (raw density) lives in lane 0
        const float raw = __int_as_float(
            __builtin_amdgcn_readfirstlane(__float_as_int(val)));
        densAcc += 1.0f - expf(-delta * fmaxf(raw, 0.0f));
    }

    if (lane == 0)
        outDens[p] = densAcc * (1.0f / 3.0f);
}

// ---------------------------------------------------------------------------
extern "C" void kernel_launch(void* const* d_in, const int* in_sizes, int n_in,
                              void* d_out, int out_size, void* d_ws, size_t ws_size,
                              hipStream_t stream)
{
    const float* points = (const float*)d_in[0];
    const float* fxy    = (const float*)d_in[1];
    const float* fyz    = (const float*)d_in[2];
    const float* fxz    = (const float*)d_in[3];
    const float* depth  = (const float*)d_in[4];

    float* outDens = (float*)d_out;              // (B,H,W,N,1) = 2,097,152 floats
    float* outCol  = outDens + NPTS;             // (B,H,W,N,48)

    const size_t needWS = (size_t)6 * PLANEF * sizeof(float);  // 24 MB
    const dim3 blkMain(256);
    const dim3 grdMain(NPTS / 8);                // 1 wave per point, 8 waves/block

    if (ws_size >= needWS) {
        float* wsT = (float*)d_ws;
        tp_transpose_tdm<<<dim3(6 * 256), dim3(256), 0, stream>>>(fxy, fyz, fxz, wsT);
        tp_main<true><<<grdMain, blkMain, 0, stream>>>(points, depth, fxy, fyz, fxz,
                                                       wsT, outDens, outCol);
    } else {
        tp_main<false><<<grdMain, blkMain, 0, stream>>>(points, depth, fxy, fyz, fxz,
                                                        nullptr, outDens, outCol);
    }
}